// QueryClippingLayer_37761352466457
// MI455X (gfx1250) — compile-verified
//
#include <hip/hip_runtime.h>
#include <stdint.h>

// ---------------------------------------------------------------------------
// Problem constants (from reference setup_inputs)
// ---------------------------------------------------------------------------
#define BATCHES 8
#define CH 64
#define HH 200
#define WW 704
#define NPB (CH * HH * WW)            // 9,011,200 elements per batch
#define N4 (NPB / 4)                  // 2,252,800 float4 per batch
#define TILE_F4 512                   // one float4 per thread per tile (512 thr)
#define TILES_PB (N4 / TILE_F4)       // 4400 tiles per batch
#define WG_PER_BATCH 44               // 4400 = 44 * 100 exactly
#define TILES_PER_WG (TILES_PB / WG_PER_BATCH)  // 100
#define HIST_BINS 65536               // 16-bit radix level
#define HIST_BYTES (HIST_BINS * 4)    // 256 KB LDS histogram (CDNA5: 320KB/WGP)
#define STAGE_DEPTH 4                 // async pipeline depth per wave
#define STAGE_BYTES (STAGE_DEPTH * TILE_F4 * 16)  // 32 KB staging
#define LDS_BYTES (HIST_BYTES + STAGE_BYTES)      // 294,912 B < 320 KB

// Native clang vector (accepted by __builtin_nontemporal_*; b128 accesses).
typedef __attribute__((ext_vector_type(4))) float v4f;

// Monotonic sortable key: larger float -> larger unsigned key (total order).
__device__ __forceinline__ unsigned sortkey(float f) {
  unsigned u = __float_as_uint(f);
  return (u & 0x80000000u) ? ~u : (u | 0x80000000u);
}

// ---------------------------------------------------------------------------
// Zero the global histogram scratch (lives in d_out's first 4 MB).
// ---------------------------------------------------------------------------
extern "C" __global__ void qcl_zero(unsigned* __restrict__ p, int n) {
  int i = blockIdx.x * blockDim.x + threadIdx.x;
  if (i < n) p[i] = 0u;
}

// ---------------------------------------------------------------------------
// Streaming radix-histogram pass.
//  mode 0: bin = key >> 16            (level 1, all elements)
//  mode 1: bin = key & 0xFFFF, only elements whose high-16 == sel[b]
// Data is streamed with CDNA5 async global->LDS loads (ASYNCcnt), 4-deep
// pipeline per wave; histogram is privatized in a 256 KB LDS array. The
// 256 KB-per-WG LDS footprint caps occupancy at ~1 WG/WGP, so latency hiding
// comes from the deep per-wave async pipeline rather than wave count.
// ---------------------------------------------------------------------------
extern "C" __global__ __launch_bounds__(512)
void qcl_hist(const v4f* __restrict__ q4, unsigned* __restrict__ ghist,
              const unsigned* __restrict__ sel, int mode) {
  extern __shared__ unsigned char smem[];
  unsigned* hist = (unsigned*)smem;
  const int tid = threadIdx.x;
  const int b = blockIdx.x / WG_PER_BATCH;
  const int wgIn = blockIdx.x % WG_PER_BATCH;

  for (int i = tid; i < HIST_BINS; i += 512) hist[i] = 0u;
  __syncthreads();

  const unsigned selv = (mode != 0) ? sel[b] : 0u;
  const size_t base4 =
      (size_t)b * N4 + (size_t)wgIn * (TILES_PER_WG * TILE_F4);

  auto issue = [&](int t) {
    unsigned long long g =
        (unsigned long long)(const void*)(q4 + base4 + (size_t)t * TILE_F4 + tid);
    unsigned lds =
        (unsigned)(HIST_BYTES + ((t & (STAGE_DEPTH - 1)) * TILE_F4 + tid) * 16);
    // CDNA5 async DMA: global memory -> LDS, tracked by ASYNCcnt.
    asm volatile("global_load_async_to_lds_b128 %0, %1, off"
                 :: "v"(lds), "v"(g) : "memory");
  };
  auto process = [&](int t) {
    const v4f v = *(const v4f*)(smem + HIST_BYTES +
        ((t & (STAGE_DEPTH - 1)) * TILE_F4 + tid) * 16);
#pragma unroll
    for (int j = 0; j < 4; ++j) {
      unsigned key = sortkey(v[j]);
      if (mode == 0) {
        atomicAdd(&hist[key >> 16], 1u);
      } else if ((key >> 16) == selv) {
        atomicAdd(&hist[key & 0xFFFFu], 1u);
      }
    }
  };

  issue(0); issue(1); issue(2); issue(3);
  for (int t = 0; t < TILES_PER_WG - STAGE_DEPTH; ++t) {
    // Async loads complete in order: <=3 outstanding => oldest tile landed.
    asm volatile("s_wait_asynccnt 0x3" ::: "memory");
    process(t);
    issue(t + STAGE_DEPTH);
  }
  asm volatile("s_wait_asynccnt 0x0" ::: "memory");
  process(TILES_PER_WG - 4); process(TILES_PER_WG - 3);
  process(TILES_PER_WG - 2); process(TILES_PER_WG - 1);

  __syncthreads();
  // Flush: skip empty bins (normal data occupies few thousand of 64K bins).
  unsigned* g = ghist + (size_t)b * HIST_BINS;
  for (int i = tid; i < HIST_BINS; i += 512) {
    unsigned v = hist[i];
    if (v) atomicAdd(&g[i], v);
  }
}

// ---------------------------------------------------------------------------
// Descending scan of a 65536-bin histogram: find bin of the k-th largest.
//  mode 0: k from budget; writes sel_out[b], rem_out[b]
//  mode 1: k = rem_in[b]; writes thr_out[b] = (sel_in<<16)|bin, rem_out[b],
//          and zeroes tie_out[b] for the mask pass.
// ---------------------------------------------------------------------------
extern "C" __global__ __launch_bounds__(256)
void qcl_scan(const unsigned* __restrict__ ghist, const int* __restrict__ budget,
              const unsigned* __restrict__ rem_in,
              const unsigned* __restrict__ sel_in,
              unsigned* sel_out, unsigned* rem_out,
              unsigned* thr_out, unsigned* tie_out, int mode) {
  __shared__ unsigned cs[256];
  const int b = blockIdx.x;
  const int tid = threadIdx.x;
  const unsigned* h = ghist + (size_t)b * HIST_BINS;

  unsigned s = 0;
  const int cbase = tid * 256;
  for (int i = 0; i < 256; ++i) s += h[cbase + i];
  cs[tid] = s;
  __syncthreads();

  if (tid == 0) {
    unsigned long long total = 0;
    for (int j = 0; j < 256; ++j) total += cs[j];
    long long k;
    if (mode == 0) {
      long long bud = (long long)(*budget);
      k = (bud <= 1) ? bud * (long long)(HH * WW) : bud;  // ratio vs absolute
    } else {
      k = (long long)rem_in[b];
    }
    if (k < 1) k = 1;
    if ((unsigned long long)k > total) k = (long long)total;  // k>=n => all-ones

    unsigned long long cum = 0;
    unsigned selbin = 0, remv = 1;
    for (int j = 255; j >= 0; --j) {
      if (cum + cs[j] >= (unsigned long long)k) {
        for (int i = 255; i >= 0; --i) {
          unsigned v = h[j * 256 + i];
          if (cum + v >= (unsigned long long)k) {
            selbin = (unsigned)(j * 256 + i);
            remv = (unsigned)((unsigned long long)k - cum);
            break;
          }
          cum += v;
        }
        break;
      }
      cum += cs[j];
    }
    if (mode == 0) {
      sel_out[b] = selbin;
      rem_out[b] = remv;
    } else {
      thr_out[b] = (sel_in[b] << 16) | selbin;
      rem_out[b] = remv;
      tie_out[b] = 0u;  // reset tie counter every call (deterministic)
    }
  }
}

// ---------------------------------------------------------------------------
// Final pass: mask = key > thr ? 1 : (key == thr ? first-rem ties : 0)
// Terminal 288 MB read + 288 MB write stream: non-temporal loads/stores so
// this no-reuse stream does not thrash the 192 MB L2 (which still holds part
// of the input from the level-2 pass). thr/rem are block-uniform -> s_load.
// ---------------------------------------------------------------------------
extern "C" __global__ __launch_bounds__(256)
void qcl_mask(const v4f* __restrict__ q4, v4f* __restrict__ o4,
              const unsigned* __restrict__ thr, const unsigned* __restrict__ rem,
              unsigned* __restrict__ tie) {
  const size_t g = (size_t)blockIdx.x * 256 + threadIdx.x;
  // N4 % 256 == 0: every block lies inside a single batch -> b is uniform.
  const int b = (int)((size_t)blockIdx.x * 256 / N4);
  const v4f v = __builtin_nontemporal_load(&q4[g]);
  const unsigned T = thr[b];
  const unsigned R = rem[b];
  v4f out;
#pragma unroll
  for (int j = 0; j < 4; ++j) {
    unsigned key = sortkey(v[j]);
    float m = 0.0f;
    if (key > T) {
      m = 1.0f;
    } else if (key == T) {
      m = (atomicAdd(&tie[b], 1u) < R) ? 1.0f : 0.0f;
    }
    out[j] = m;
  }
  __builtin_nontemporal_store(out, &o4[g]);
}

// ---------------------------------------------------------------------------
// Host-side launch sequence (all on `stream`, graph-capture safe).
// Big histogram scratch lives inside d_out (fully overwritten by qcl_mask);
// only 160 B of control state uses d_ws.
// ---------------------------------------------------------------------------
extern "C" void kernel_launch(void* const* d_in, const int* in_sizes, int n_in,
                              void* d_out, int out_size, void* d_ws, size_t ws_size,
                              hipStream_t stream) {
  const v4f* q4 = (const v4f*)d_in[0];
  const int* budget = (const int*)d_in[1];

  unsigned* hist1 = (unsigned*)d_out;                       // 2 MB
  unsigned* hist2 = hist1 + (size_t)BATCHES * HIST_BINS;    // 2 MB
  unsigned* ws = (unsigned*)d_ws;
  unsigned* sel1 = ws + 0;   // [8]
  unsigned* rem1 = ws + 8;   // [8]
  unsigned* thr  = ws + 16;  // [8]
  unsigned* rem2 = ws + 24;  // [8]
  unsigned* tie  = ws + 32;  // [8]

  const int histWords = 2 * BATCHES * HIST_BINS;  // 1,048,576 u32
  qcl_zero<<<(histWords + 255) / 256, 256, 0, stream>>>(hist1, histWords);

  // Level 1: top-16-bit radix histogram (288 MB streamed via async->LDS).
  qcl_hist<<<BATCHES * WG_PER_BATCH, 512, LDS_BYTES, stream>>>(
      q4, hist1, nullptr, 0);
  qcl_scan<<<BATCHES, 256, 0, stream>>>(
      hist1, budget, nullptr, nullptr, sel1, rem1, nullptr, nullptr, 0);

  // Level 2: low-16-bit radix histogram within the selected bucket.
  qcl_hist<<<BATCHES * WG_PER_BATCH, 512, LDS_BYTES, stream>>>(
      q4, hist2, sel1, 1);
  qcl_scan<<<BATCHES, 256, 0, stream>>>(
      hist2, budget, rem1, sel1, nullptr, rem2, thr, tie, 1);

  // Final mask write (overwrites the histogram scratch region too).
  const size_t total4 = (size_t)BATCHES * N4;  // 18,022,400
  qcl_mask<<<(unsigned)(total4 / 256), 256, 0, stream>>>(
      q4, (v4f*)d_out, thr, rem2, tie);
}